// NerfAccSampler_47596827574453
// MI455X (gfx1250) — compile-verified
//
#include <hip/hip_runtime.h>
#include <stdint.h>

#define TILE 256  // blockDim and LDS staging chunk (samples)

typedef unsigned int u32x4 __attribute__((ext_vector_type(4)));
typedef int          i32x4 __attribute__((ext_vector_type(4)));
typedef int          i32x8 __attribute__((ext_vector_type(8)));

// ---------------------------------------------------------------------------
// TDM: stage n_elems consecutive f32 from global memory into LDS (1-D tile).
// D# layout per CDNA5 ISA §8 (CDNA5 ISA):
//   group0: [1:0]=count=1, [63:32]=lds_addr, [120:64]=global_addr, [127:126]=2
//   group1: wg_mask=0 (not in cluster), data_size=2 (4B),
//           tensor_dim0 = tile_dim0 = n, tensor_dim1 = tile_dim1 = 1,
//           tensor_dim0_stride = n
//   remaining groups: zero (dims 2..4 unused / benign)
// This toolchain (clang-23 / amdgpu-toolchain) uses the 6-arg builtin:
//   (u32x4, i32x8, i32x4, i32x4, i32x8, i32 cpol)
// ---------------------------------------------------------------------------
__device__ __forceinline__ void tdm_load_row_to_lds(const void* gptr,
                                                    uint32_t lds_byte_off,
                                                    int n_elems) {
  const uint64_t ga   = (uint64_t)(uintptr_t)gptr;
  const uint32_t dim0 = (uint32_t)n_elems;

  u32x4 g0;
  g0[0] = 1u;                                   // count=1 (valid user D#)
  g0[1] = lds_byte_off;                         // lds_addr (bytes)
  g0[2] = (uint32_t)ga;                         // global_addr[31:0]
  g0[3] = (uint32_t)((ga >> 32) & 0x01FFFFFFull) | (2u << 30); // addr[56:32] | type=2

  i32x8 g1;
  g1[0] = (int)(2u << 16);                      // wg_mask=0, data_size=4B
  g1[1] = (int)((dim0 & 0xFFFFu) << 16);        // tensor_dim0[15:0] @ bits 63:48
  g1[2] = (int)(((dim0 >> 16) & 0xFFFFu)        // tensor_dim0[31:16]
                | (1u << 16));                  // tensor_dim1 = 1
  g1[3] = (int)((dim0 & 0xFFFFu) << 16);        // tile_dim0 = n (bits 127:112)
  g1[4] = 1;                                    // tile_dim1 = 1, tile_dim2 = 0
  g1[5] = (int)dim0;                            // tensor_dim0_stride[31:0]
  g1[6] = 0;                                    // stride0[47:32]=0, stride1 lo=0
  g1[7] = 0;

  i32x4 gz4; gz4[0] = 0; gz4[1] = 0; gz4[2] = 0; gz4[3] = 0;
  i32x8 gz8; gz8[0] = 0; gz8[1] = 0; gz8[2] = 0; gz8[3] = 0;
             gz8[4] = 0; gz8[5] = 0; gz8[6] = 0; gz8[7] = 0;

  __builtin_amdgcn_tensor_load_to_lds(g0, g1, gz4, gz4, gz8, 0);
}

// ---------------------------------------------------------------------------
// Phase 1: offsets[r] = lower_bound(ray_indices, r), r in [0, N]  (sorted input)
// ---------------------------------------------------------------------------
__global__ void offsets_kernel(const int* __restrict__ ray_indices,
                               int* __restrict__ offsets, int N, int T) {
  const int r = blockIdx.x * blockDim.x + threadIdx.x;
  if (r > N) return;
  int lo = 0, hi = T;
  while (lo < hi) {
    const int mid = (lo + hi) >> 1;
    if (ray_indices[mid] < r) lo = mid + 1; else hi = mid;
  }
  offsets[r] = lo;
}

// ---------------------------------------------------------------------------
// Phase 2: one block per ray. TDM-stage its contiguous t_start/t_end segment
// into LDS chunk-by-chunk, emit xyz_sampled (T,4) and the padded (N,M) arrays.
// ---------------------------------------------------------------------------
__global__ void __launch_bounds__(TILE)
sampler_kernel(const float* __restrict__ rays_chunk,
               const float* __restrict__ t_starts,
               const float* __restrict__ t_ends,
               const int*   __restrict__ offsets,
               float* __restrict__ out,
               int N, int T, int M) {
  __shared__ alignas(16) float lds_ts[TILE];
  __shared__ alignas(16) float lds_te[TILE];
  __shared__ float ray[6];

  const int r   = blockIdx.x;
  const int tid = threadIdx.x;

  const long long NM = (long long)N * (long long)M;
  float* out_xyz = out;                       // (T,4)
  float* out_rv  = out + 4ll * (long long)T;  // (N,M) bool -> 0/1 f32
  float* out_z   = out_rv + NM;               // (N,M)
  float* out_d   = out_z  + NM;               // (N,M)
  float* out_wv  = out_d  + NM;               // (N,)  all 1.0
  float* out_xw  = out_wv + N;                // (N,M,4)

  if (tid < 6)  ray[tid] = rays_chunk[(long long)r * 6 + tid];
  if (tid == 0) out_wv[r] = 1.0f;

  const int s_off = offsets[r];
  const int s_end = offsets[r + 1];
  const int cnt   = s_end - s_off;

  __syncthreads();  // ray[] visible to all waves

  const float ox = ray[0], oy = ray[1], oz = ray[2];
  const float dx = ray[3], dy = ray[4], dz = ray[5];

  const uint32_t lds_ts_off = (uint32_t)(uintptr_t)&lds_ts[0];
  const uint32_t lds_te_off = (uint32_t)(uintptr_t)&lds_te[0];

  const int nchunks = (cnt + TILE - 1) / TILE;  // block-uniform
  for (int c = 0; c < nchunks; ++c) {
    const int base    = s_off + c * TILE;
    const int chunk_n = min(TILE, s_end - base);

    if (tid == 0) {  // wave 0 issues the async tensor DMAs (EXEC ignored by TDM)
      tdm_load_row_to_lds(t_starts + base, lds_ts_off, chunk_n);
      tdm_load_row_to_lds(t_ends   + base, lds_te_off, chunk_n);
      __builtin_amdgcn_s_wait_tensorcnt(0);
    }
    __syncthreads();  // LDS tiles ready for all 8 waves

    if (tid < chunk_n) {
      const float a  = lds_ts[tid];
      const float b  = lds_te[tid];
      const float pz = 0.5f * (a + b);
      const float pd = b - a;
      const int   t   = base + tid;
      const int   pos = t - s_off;

      float4 xs;
      xs.x = fmaf(pz, dx, ox);
      xs.y = fmaf(pz, dy, oy);
      xs.z = fmaf(pz, dz, oz);
      xs.w = 0.0f;
      reinterpret_cast<float4*>(out_xyz)[t] = xs;  // global_store_b128

      if (pos < M) {  // matches JAX dropped-OOB scatter semantics
        const long long idx = (long long)r * M + pos;
        out_rv[idx] = 1.0f;
        out_z[idx]  = pz;
        out_d[idx]  = pd;
        reinterpret_cast<float4*>(out_xw)[idx] = xs;
      }
    }
    __syncthreads();  // protect LDS tiles before next chunk overwrites
  }

  // zero-fill padded tail [cnt, M)  (ray_valid = 0 there)
  float4 z4; z4.x = z4.y = z4.z = z4.w = 0.0f;
  for (int p = cnt + tid; p < M; p += TILE) {
    const long long idx = (long long)r * M + p;
    out_rv[idx] = 0.0f;
    out_z[idx]  = 0.0f;
    out_d[idx]  = 0.0f;
    reinterpret_cast<float4*>(out_xw)[idx] = z4;
  }
}

// ---------------------------------------------------------------------------
extern "C" void kernel_launch(void* const* d_in, const int* in_sizes, int n_in,
                              void* d_out, int out_size, void* d_ws, size_t ws_size,
                              hipStream_t stream) {
  const float* rays_chunk  = (const float*)d_in[0];   // (N,6) f32
  const int*   ray_indices = (const int*)  d_in[1];   // (T,)  i32 sorted
  const float* t_starts    = (const float*)d_in[2];   // (T,)  f32
  const float* t_ends      = (const float*)d_in[3];   // (T,)  f32

  const int N = in_sizes[0] / 6;
  const int T = in_sizes[1];
  if (N <= 0 || T <= 0) return;

  // out_size = 4T + 7*N*M + N  ->  M (max_samps) recovered host-side
  const long long M = ((long long)out_size - 4ll * T - N) / (7ll * N);

  int* offsets = (int*)d_ws;  // (N+1) ints of scratch

  offsets_kernel<<<(N + 1 + 255) / 256, 256, 0, stream>>>(ray_indices, offsets, N, T);
  sampler_kernel<<<N, TILE, 0, stream>>>(rays_chunk, t_starts, t_ends, offsets,
                                         (float*)d_out, N, T, (int)M);
}